// IntraModalEnhancer_48661979463836
// MI455X (gfx1250) — compile-verified
//
#include <hip/hip_runtime.h>
#include <math.h>

#define DD 1024
#define BB 256

typedef float v2f __attribute__((ext_vector_type(2)));
typedef float v8f __attribute__((ext_vector_type(8)));
typedef int   v4i __attribute__((ext_vector_type(4)));

#if __has_builtin(__builtin_amdgcn_global_load_async_to_lds_b128) && \
    __has_builtin(__builtin_amdgcn_s_wait_asynccnt)
#define HAVE_ASYNC_LDS 1
typedef __attribute__((address_space(1))) v4i* gv4i_p;   // global src
typedef __attribute__((address_space(3))) v4i* lv4i_p;   // LDS dst
#endif

// ---------------------------------------------------------------------------
// Kernel 1: QKV projections with FP32 WMMA (V_WMMA_F32_16X16X4_F32).
//   out[b,n] = sum_d x[b,d] * W[n,d] + bias[n]      (torch Linear: x @ W.T + b)
// One wave computes one 16x16 output tile. 4-way k-split accumulators give
// four independent WMMA dependency chains (D feeds only its own C), so the
// matrix pipe stays busy even at 1 wave/SIMD; partial sums are exact fp32
// per-lane adds at the end.
// A layout (16x4 f32): lanes 0-15 hold M=lane, VGPR0=K0/K2, VGPR1=K1/K3
//                      (lanes 16-31 carry K=2,3) -> each lane loads a float2.
// B layout (4x16 f32): B[k][n] = W[n][k] -> identical float2 shape per lane.
// C/D layout: VGPR r -> row M=r (lanes 0-15) and M=r+8 (lanes 16-31).
// ---------------------------------------------------------------------------
__global__ __launch_bounds__(32)
void qkv_wmma_kernel(const float* __restrict__ x,
                     const float* __restrict__ Wq, const float* __restrict__ bq,
                     const float* __restrict__ Wk, const float* __restrict__ bk,
                     const float* __restrict__ Wv, const float* __restrict__ bv,
                     float* __restrict__ ws)
{
    const int lane = threadIdx.x;     // 0..31, wave32
    const int lr   = lane & 15;
    const int hi   = lane >> 4;       // 0: K offsets {0,1}; 1: K offsets {2,3}
    const int rowBase = blockIdx.x * 16;   // batch rows (M)
    const int colBase = blockIdx.y * 16;   // output features (N)
    const int which   = blockIdx.z;        // 0=Q, 1=K, 2=V

    const float* W    = (which == 0) ? Wq : (which == 1) ? Wk : Wv;
    const float* bias = (which == 0) ? bq : (which == 1) ? bk : bv;
    float* out = ws + (size_t)which * BB * DD;

    const float* aRow = x + (size_t)(rowBase + lr) * DD + 2 * hi;
    const float* bRow = W + (size_t)(colBase + lr) * DD + 2 * hi;

    // Bias depends only on N -> identical value in all 8 accumulator VGPRs.
    const float bb = bias[colBase + lr];
    v8f c0 = { bb, bb, bb, bb, bb, bb, bb, bb };
    v8f c1 = { 0, 0, 0, 0, 0, 0, 0, 0 };
    v8f c2 = { 0, 0, 0, 0, 0, 0, 0, 0 };
    v8f c3 = { 0, 0, 0, 0, 0, 0, 0, 0 };

    for (int k = 0; k < DD; k += 16) {
        v2f a0 = *(const v2f*)(aRow + k);
        v2f b0 = *(const v2f*)(bRow + k);
        v2f a1 = *(const v2f*)(aRow + k + 4);
        v2f b1 = *(const v2f*)(bRow + k + 4);
        v2f a2 = *(const v2f*)(aRow + k + 8);
        v2f b2 = *(const v2f*)(bRow + k + 8);
        v2f a3 = *(const v2f*)(aRow + k + 12);
        v2f b3 = *(const v2f*)(bRow + k + 12);
        if (k + 128 < DD) {                      // uniform branch, EXEC untouched
            __builtin_prefetch(aRow + k + 128, 0, 0);   // global_prefetch_b8
            __builtin_prefetch(bRow + k + 128, 0, 0);
        }
        // 8 args: (neg_a, A, neg_b, B, c_mod, C, reuse_a, reuse_b)
        c0 = __builtin_amdgcn_wmma_f32_16x16x4_f32(false, a0, false, b0, (short)0, c0, false, false);
        c1 = __builtin_amdgcn_wmma_f32_16x16x4_f32(false, a1, false, b1, (short)0, c1, false, false);
        c2 = __builtin_amdgcn_wmma_f32_16x16x4_f32(false, a2, false, b2, (short)0, c2, false, false);
        c3 = __builtin_amdgcn_wmma_f32_16x16x4_f32(false, a3, false, b3, (short)0, c3, false, false);
    }

    const v8f c = (c0 + c1) + (c2 + c3);   // exact fp32 partial-sum merge

    float* oRow = out + (size_t)colBase + lr;
    #pragma unroll
    for (int r = 0; r < 8; ++r)
        oRow[(size_t)(rowBase + r + 8 * hi) * DD] = c[r];
}

// ---------------------------------------------------------------------------
// Kernel 2: fused rank-1 attention + residual + RMSNorm. One block per batch
// row. scores[b,i,:] = q_i * K[b,:]  (rank-1), so the softmax max is simply
// q_i*Kmax (q>=0) or q_i*Kmin (q<0): a single D-length min/max reduction
// replaces the D^2 max pass, and the 1 GB scores tensor is never materialized.
// K/V rows are staged into LDS via GLOBAL_LOAD_ASYNC_TO_LDS_B128 (ASYNCcnt);
// each thread then streams 4 i-values so every sK[j]/sV[j] LDS broadcast read
// feeds 4 exp-FMA chains.
// ---------------------------------------------------------------------------
__global__ __launch_bounds__(256)
void attn_rms_kernel(const float* __restrict__ x,
                     const float* __restrict__ ws,
                     const float* __restrict__ scale,
                     const float* __restrict__ norm_w,
                     float* __restrict__ out)
{
    __shared__ float sK[DD];
    __shared__ float sV[DD];
    __shared__ float redA[256];
    __shared__ float redB[256];

    const int b   = blockIdx.x;
    const int tid = threadIdx.x;
    const float* Q = ws;
    const float* K = ws + (size_t)BB * DD;
    const float* V = ws + 2 * (size_t)BB * DD;

    float kmax = -INFINITY, kmin = INFINITY;

#ifdef HAVE_ASYNC_LDS
    {
        // 256 threads x 16B = the full 4 KB row in one async b128 per array.
        const int i4 = tid * 4;
        __builtin_amdgcn_global_load_async_to_lds_b128(
            (gv4i_p)(K + (size_t)b * DD + i4),
            (lv4i_p)(&sK[i4]), 0, 0);
        __builtin_amdgcn_global_load_async_to_lds_b128(
            (gv4i_p)(V + (size_t)b * DD + i4),
            (lv4i_p)(&sV[i4]), 0, 0);
        __builtin_amdgcn_s_wait_asynccnt(0);   // s_wait_asynccnt 0
        __syncthreads();                       // cross-wave visibility
    }
    for (int i = tid; i < DD; i += 256) {
        const float kv = sK[i];
        kmax = fmaxf(kmax, kv);
        kmin = fminf(kmin, kv);
    }
#else
    for (int i = tid; i < DD; i += 256) {
        const float kv = K[(size_t)b * DD + i];
        const float vv = V[(size_t)b * DD + i];
        sK[i] = kv;  sV[i] = vv;
        kmax = fmaxf(kmax, kv);
        kmin = fminf(kmin, kv);
    }
#endif

    redA[tid] = kmax;  redB[tid] = kmin;
    __syncthreads();
    for (int s = 128; s > 0; s >>= 1) {
        if (tid < s) {
            redA[tid] = fmaxf(redA[tid], redA[tid + s]);
            redB[tid] = fminf(redB[tid], redB[tid + s]);
        }
        __syncthreads();
    }
    kmax = redA[0];
    kmin = redB[0];
    __syncthreads();   // redA reused below

    const float invScale = 1.0f / scale[0];

    float q[4], m[4], num[4], den[4];
    #pragma unroll
    for (int t = 0; t < 4; ++t) {
        const int i = tid + t * 256;
        q[t]   = Q[(size_t)b * DD + i] * invScale;
        m[t]   = (q[t] >= 0.0f) ? q[t] * kmax : q[t] * kmin;  // exact softmax max
        num[t] = 0.0f;
        den[t] = 0.0f;
    }

    // Transcendental-bound core: 4 exp chains per LDS broadcast pair.
    #pragma unroll 4
    for (int j = 0; j < DD; ++j) {
        const float kj = sK[j];
        const float vj = sV[j];
        #pragma unroll
        for (int t = 0; t < 4; ++t) {
            const float e = __expf(fmaf(q[t], kj, -m[t]));  // v_exp_f32
            den[t] += e;
            num[t] = fmaf(e, vj, num[t]);
        }
    }

    // Residual + RMSNorm over the D axis (block-wide sum of squares).
    float h[4];
    float ss = 0.0f;
    #pragma unroll
    for (int t = 0; t < 4; ++t) {
        const int i = tid + t * 256;
        h[t] = num[t] / den[t] + x[(size_t)b * DD + i];
        ss   = fmaf(h[t], h[t], ss);
    }
    redA[tid] = ss;
    __syncthreads();
    for (int s = 128; s > 0; s >>= 1) {
        if (tid < s) redA[tid] += redA[tid + s];
        __syncthreads();
    }
    const float rinv = 1.0f / sqrtf(redA[0] * (1.0f / (float)DD) + 1e-6f);

    #pragma unroll
    for (int t = 0; t < 4; ++t) {
        const int i = tid + t * 256;
        out[(size_t)b * DD + i] = h[t] * rinv * norm_w[i];
    }
}

// ---------------------------------------------------------------------------
extern "C" void kernel_launch(void* const* d_in, const int* in_sizes, int n_in,
                              void* d_out, int out_size, void* d_ws, size_t ws_size,
                              hipStream_t stream) {
    const float* x      = (const float*)d_in[0];
    const float* Wq     = (const float*)d_in[1];
    const float* bq     = (const float*)d_in[2];
    const float* Wk     = (const float*)d_in[3];
    const float* bk     = (const float*)d_in[4];
    const float* Wv     = (const float*)d_in[5];
    const float* bv     = (const float*)d_in[6];
    const float* scale  = (const float*)d_in[7];
    const float* norm_w = (const float*)d_in[8];
    float* ws  = (float*)d_ws;     // Q | K | V, each BB*DD floats (3 MB total)
    float* out = (float*)d_out;

    dim3 gridQKV(BB / 16, DD / 16, 3);   // 16 x 64 x 3 tiles, 1 wave each
    qkv_wmma_kernel<<<gridQKV, dim3(32, 1, 1), 0, stream>>>(
        x, Wq, bq, Wk, bk, Wv, bv, ws);

    attn_rms_kernel<<<dim3(BB, 1, 1), dim3(256, 1, 1), 0, stream>>>(
        x, ws, scale, norm_w, out);
}